// QuantumNATEnhanced_65481071398686
// MI455X (gfx1250) — compile-verified
//
#include <hip/hip_runtime.h>
#include <math.h>

// ---------------------------------------------------------------------------
// QuantumNAT fused pipeline for gfx1250 (MI455X, wave32).
//  K1: pool+encoder as f32 WMMA GEMM (V_WMMA_F32_16X16X4_F32) + 4-qubit
//      statevector sim in registers + deterministic per-block BN partials.
//  K2: deterministic reduce of block partials (1 block).
//  K3: BatchNorm apply.
// Memory-bound: 151MB read of x -> ~6.5us floor at 23.3 TB/s.
//
// Round-2 changes vs round-1:
//  * sWt gets a 5th all-zero column; lanes with n>=4 read it, so the inner
//    K-loop has NO exec-mask divergence (no s_and_saveexec per chunk).
//  * Uniform variational cos/sin (24 pairs) computed once per block into LDS
//    instead of 24 sinf/cosf expansions per thread.
// ---------------------------------------------------------------------------

typedef __attribute__((ext_vector_type(2))) float v2f;
typedef __attribute__((ext_vector_type(8))) float v8f;

#define THREADS 256
#define ROWS_PER_BLOCK 256   // samples per block (8 waves x 32 samples)
#define KDIM 576             // 24*24 pixels per sample
#define KCHUNKS (KDIM / 4)   // 144 WMMA K-steps
#define NCOLS 5              // 4 real encoder columns + 1 zero column

// ---------------- quantum gate helpers (fully unrolled, mask const-prop) ----
__device__ __forceinline__ void gate_rx(float* re, float* im, int mask, float c, float s) {
#pragma unroll
  for (int i = 0; i < 16; ++i) {
    if ((i & mask) == 0) {
      int j = i | mask;
      float r0 = re[i], i0 = im[i], r1 = re[j], i1 = im[j];
      re[i] = c * r0 + s * i1;  im[i] = c * i0 - s * r1;
      re[j] = c * r1 + s * i0;  im[j] = c * i1 - s * r0;
    }
  }
}
__device__ __forceinline__ void gate_ry(float* re, float* im, int mask, float c, float s) {
#pragma unroll
  for (int i = 0; i < 16; ++i) {
    if ((i & mask) == 0) {
      int j = i | mask;
      float r0 = re[i], i0 = im[i], r1 = re[j], i1 = im[j];
      re[i] = c * r0 - s * r1;  im[i] = c * i0 - s * i1;
      re[j] = s * r0 + c * r1;  im[j] = s * i0 + c * i1;
    }
  }
}
__device__ __forceinline__ void gate_rz(float* re, float* im, int mask, float c, float s) {
#pragma unroll
  for (int i = 0; i < 16; ++i) {
    if ((i & mask) == 0) {
      int j = i | mask;
      float r0 = re[i], i0 = im[i], r1 = re[j], i1 = im[j];
      // a0 *= (c - i s); a1 *= (c + i s)
      re[i] = c * r0 + s * i0;  im[i] = c * i0 - s * r0;
      re[j] = c * r1 - s * i1;  im[j] = c * i1 + s * r1;
    }
  }
}
__device__ __forceinline__ void gate_cnot(float* re, float* im, int cmask, int tmask) {
#pragma unroll
  for (int i = 0; i < 16; ++i) {
    if ((i & cmask) && !(i & tmask)) {
      int j = i | tmask;
      float tr = re[i], ti = im[i];
      re[i] = re[j];  im[i] = im[j];
      re[j] = tr;     im[j] = ti;
    }
  }
}

// ---------------- kernel 1: GEMM(WMMA) + circuit + BN partials --------------
__global__ __launch_bounds__(THREADS)
void qnat_main_kernel(const float* __restrict__ x,
                      const float* __restrict__ enc_w,
                      const float* __restrict__ enc_b,
                      const float* __restrict__ vp,
                      float* __restrict__ zbuf,
                      float* __restrict__ partials) {
  __shared__ float sWt[NCOLS * KDIM];        // W^T: [n][k]; column 4 = zeros
  __shared__ float sAng[ROWS_PER_BLOCK * 4]; // angles per sample
  __shared__ float sCos[24], sSin[24];       // uniform variational trig
  __shared__ float sRed[8][8];               // per-wave BN partials

  const int tid  = threadIdx.x;
  const int wave = tid >> 5;
  const int lane = tid & 31;
  const int blockBase = blockIdx.x * ROWS_PER_BLOCK;

  // Build W^T in LDS: W[k,n] = enc_w[n, poolidx(k)] / 36 ; column 4 is zero.
  for (int i = tid; i < NCOLS * KDIM; i += THREADS) {
    int n = i / KDIM;
    int k = i - n * KDIM;
    int pi = (k / 144) * 4 + ((k % 24) / 6);   // (row/6)*4 + (col/6)
    sWt[i] = (n < 4) ? enc_w[n * 16 + pi] * (1.0f / 36.0f) : 0.0f;
  }
  // Uniform variational trig, once per block.
  if (tid < 24) {
    float h = 0.5f * vp[tid];
    sCos[tid] = cosf(h);
    sSin[tid] = sinf(h);
  }
  __syncthreads();

  // ---- WMMA fragment coordinates (CDNA5 ISA layouts) ----
  const int m     = lane & 15;   // A row / B-D column
  const int khalf = lane >> 4;   // 0: K={0,1}, 1: K={2,3}
  const int row0  = blockBase + wave * 32;

  const float* xr0 = x + (size_t)(row0 + m) * KDIM + khalf * 2;
  const float* xr1 = xr0 + (size_t)16 * KDIM;
  const int ncol  = (m < 4) ? m : 4;         // n>=4 lanes -> zero column
  const float* wcol = sWt + ncol * KDIM + khalf * 2;

  __builtin_prefetch(xr0, 0, 3);   // global_prefetch_b8
  __builtin_prefetch(xr1, 0, 3);

  v8f c0 = {};
  v8f c1 = {};
#pragma unroll 8
  for (int kk = 0; kk < KCHUNKS; ++kk) {
    v2f a0 = *(const v2f*)(xr0 + kk * 4);
    v2f a1 = *(const v2f*)(xr1 + kk * 4);
    v2f b  = *(const v2f*)(wcol + kk * 4);   // unconditional ds_load_b64
    c0 = __builtin_amdgcn_wmma_f32_16x16x4_f32(false, a0, false, b,
                                               (short)0, c0, false, false);
    c1 = __builtin_amdgcn_wmma_f32_16x16x4_f32(false, a1, false, b,
                                               (short)0, c1, false, false);
  }

  // Scatter D -> per-sample angles in LDS (add enc_b)
  if (m < 4) {
    float bc = enc_b[m];
#pragma unroll
    for (int j = 0; j < 8; ++j) {
      int mr = j + 8 * khalf;
      sAng[(wave * 32 + mr) * 4 + m]      = c0[j] + bc;
      sAng[(wave * 32 + 16 + mr) * 4 + m] = c1[j] + bc;
    }
  }
  __syncthreads();

  // ---- per-sample 4-qubit statevector sim (registers only) ----
  float ang[4];
#pragma unroll
  for (int i = 0; i < 4; ++i) ang[i] = sAng[tid * 4 + i];

  float re[16], im[16];
#pragma unroll
  for (int i = 0; i < 16; ++i) { re[i] = 0.0f; im[i] = 0.0f; }
  re[0] = 1.0f;

  // encoding RX per wire (wire w -> bit mask 8>>w)
#pragma unroll
  for (int w = 0; w < 4; ++w) {
    float h = 0.5f * ang[w];
    gate_rx(re, im, 8 >> w, cosf(h), sinf(h));
  }
  // variational layers (trig from LDS, uniform -> scalar loads)
#pragma unroll
  for (int d = 0; d < 2; ++d) {
#pragma unroll
    for (int i = 0; i < 4; ++i) {
      int mask = 8 >> i;
      int base = d * 12 + i * 3;
      gate_rx(re, im, mask, sCos[base + 0], sSin[base + 0]);
      gate_ry(re, im, mask, sCos[base + 1], sSin[base + 1]);
      gate_rz(re, im, mask, sCos[base + 2], sSin[base + 2]);
    }
    gate_cnot(re, im, 8, 4);  // (0,1)
    gate_cnot(re, im, 4, 2);  // (1,2)
    gate_cnot(re, im, 2, 1);  // (2,3)
    gate_cnot(re, im, 1, 8);  // (3,0)
  }

  // Z expectations
  float z[4] = {0.0f, 0.0f, 0.0f, 0.0f};
#pragma unroll
  for (int i = 0; i < 16; ++i) {
    float p = re[i] * re[i] + im[i] * im[i];
#pragma unroll
    for (int w = 0; w < 4; ++w)
      z[w] += (i & (8 >> w)) ? -p : p;
  }

  // store z for BN pass (float4, 16B aligned)
  float4 zv = make_float4(z[0], z[1], z[2], z[3]);
  *(float4*)(zbuf + (size_t)(blockBase + tid) * 4) = zv;

  // ---- deterministic BN partials: wave shuffle -> LDS -> fixed-order sum ----
  float v[8];
#pragma unroll
  for (int k = 0; k < 4; ++k) { v[k] = z[k]; v[4 + k] = z[k] * z[k]; }
#pragma unroll
  for (int off = 16; off > 0; off >>= 1) {
#pragma unroll
    for (int k = 0; k < 8; ++k) v[k] += __shfl_down(v[k], off, 32);
  }
  if (lane == 0) {
#pragma unroll
    for (int k = 0; k < 8; ++k) sRed[wave][k] = v[k];
  }
  __syncthreads();
  if (tid < 8) {
    float a = 0.0f;
#pragma unroll
    for (int w = 0; w < 8; ++w) a += sRed[w][tid];
    partials[blockIdx.x * 8 + tid] = a;
  }
}

// ---------------- kernel 2: deterministic final reduce ----------------------
__global__ void qnat_reduce_kernel(const float* __restrict__ partials,
                                   float* __restrict__ sums, int nblocks) {
  int n = threadIdx.x;
  if (n < 8) {
    float a = 0.0f;
    for (int b = 0; b < nblocks; ++b) a += partials[b * 8 + n];
    sums[n] = a;
  }
}

// ---------------- kernel 3: BatchNorm apply ---------------------------------
__global__ __launch_bounds__(THREADS)
void qnat_bn_kernel(const float* __restrict__ zbuf,
                    const float* __restrict__ sums,
                    const float* __restrict__ gamma,
                    const float* __restrict__ beta,
                    float* __restrict__ out, int Btot) {
  int s = blockIdx.x * THREADS + threadIdx.x;
  if (s >= Btot) return;
  float invB = 1.0f / (float)Btot;
  float4 zv = *(const float4*)(zbuf + (size_t)s * 4);
  float zz[4] = {zv.x, zv.y, zv.z, zv.w};
  float4 o;
  float* op = (float*)&o;
#pragma unroll
  for (int n = 0; n < 4; ++n) {
    float mean = sums[n] * invB;
    float var  = sums[4 + n] * invB - mean * mean;
    float r    = rsqrtf(var + 1e-5f);
    op[n] = (zz[n] - mean) * r * gamma[n] + beta[n];
  }
  *(float4*)(out + (size_t)s * 4) = o;
}

// ---------------- host launcher ---------------------------------------------
extern "C" void kernel_launch(void* const* d_in, const int* in_sizes, int n_in,
                              void* d_out, int out_size, void* d_ws, size_t ws_size,
                              hipStream_t stream) {
  (void)n_in; (void)out_size; (void)ws_size;
  const float* x      = (const float*)d_in[0];
  const float* enc_w  = (const float*)d_in[1];
  const float* enc_b  = (const float*)d_in[2];
  const float* vparam = (const float*)d_in[3];
  const float* gamma  = (const float*)d_in[4];
  const float* beta   = (const float*)d_in[5];
  float* out = (float*)d_out;

  const int B = in_sizes[0] / KDIM;          // 65536
  const int nblocks = B / ROWS_PER_BLOCK;    // 256

  // ws layout (floats): [0 .. nblocks*8) partials | [nblocks*8 .. +8) sums |
  //                     [4096 ..) zbuf (B*4 floats)
  float* ws       = (float*)d_ws;
  float* partials = ws;
  float* sums     = ws + nblocks * 8;
  float* zbuf     = ws + 4096;

  qnat_main_kernel<<<nblocks, THREADS, 0, stream>>>(x, enc_w, enc_b, vparam,
                                                    zbuf, partials);
  qnat_reduce_kernel<<<1, 8, 0, stream>>>(partials, sums, nblocks);
  qnat_bn_kernel<<<nblocks, THREADS, 0, stream>>>(zbuf, sums, gamma, beta,
                                                  out, B);
}